// IntpFeat_35424890257899
// MI455X (gfx1250) — compile-verified
//
#include <hip/hip_runtime.h>

typedef float v2f __attribute__((ext_vector_type(2)));
typedef float v8f __attribute__((ext_vector_type(8)));

#define B_   4
#define N_   16384
#define M_   4096
#define C_   64
#define EPS_ 1e-8f
#define MTILES (M_ / 16)          // 256 source tiles of 16
#define WPB  8                    // waves per block (256 threads)

// ---------------------------------------------------------------------------
// Branchless top-3 (smallest) insert with index tracking.
// Maintains d0 <= d1 <= d2.
// ---------------------------------------------------------------------------
__device__ __forceinline__ void top3_insert(float v, int idx,
                                            float& d0, float& d1, float& d2,
                                            int& i0, int& i1, int& i2) {
    bool lt0 = v < d0;
    bool lt1 = v < d1;
    bool lt2 = v < d2;
    d2 = lt1 ? d1 : (lt2 ? v   : d2);
    i2 = lt1 ? i1 : (lt2 ? idx : i2);
    d1 = lt0 ? d0 : (lt1 ? v   : d1);
    i1 = lt0 ? i0 : (lt1 ? idx : i1);
    d0 = lt0 ? v   : d0;
    i0 = lt0 ? idx : i0;
}

// ---------------------------------------------------------------------------
// Prep: pack xyz2 into the WMMA B-operand (4x16, K x N) lane layout:
//   per 16-source tile, 32 lanes * 2 floats:
//     lane n      (n<16): (x_n, y_n)        -> K=0,1
//     lane n+16         : (z_n, |q_n|^2)    -> K=2,3
// ---------------------------------------------------------------------------
__global__ void IntpFeat_prep_kernel(const float* __restrict__ xyz2,
                                     v2f* __restrict__ wsB) {
    int i = blockIdx.x * blockDim.x + threadIdx.x;
    if (i >= B_ * M_) return;
    int b = i / M_;
    int s = i - b * M_;
    float x = xyz2[(size_t)i * 3 + 0];
    float y = xyz2[(size_t)i * 3 + 1];
    float z = xyz2[(size_t)i * 3 + 2];
    int tile = s >> 4;
    int col  = s & 15;
    v2f* o = wsB + (((size_t)(b * MTILES + tile)) << 5);
    v2f lo; lo.x = x; lo.y = y;
    v2f hi; hi.x = z; hi.y = x * x + y * y + z * z;
    o[col]      = lo;
    o[col + 16] = hi;
}

// ---------------------------------------------------------------------------
// Main: one wave per 16 target points. Distance scores via
// V_WMMA_F32_16X16X4_F32; per-lane running top-3; shfl_xor merge; weighted
// feature gather.
// ---------------------------------------------------------------------------
__global__ __launch_bounds__(WPB * 32)
void IntpFeat_knn_kernel(const float* __restrict__ xyz1,
                         const float* __restrict__ feat2,
                         const v2f*   __restrict__ wsB,
                         float* __restrict__ out) {
    const int lane    = threadIdx.x & 31;
    const int wave    = threadIdx.x >> 5;
    const int wglob   = blockIdx.x * WPB + wave;        // 0 .. B*N/16-1
    const int b       = wglob / (N_ / 16);
    const int n0      = (wglob - b * (N_ / 16)) * 16;

    // ---- A operand: 16x4 f32, row m = (-2x, -2y, -2z, 1) ----
    // ISA layout: lanes 0-15 hold K=0,1 ; lanes 16-31 hold K=2,3 (M = lane&15)
    const int mrow = lane & 15;
    const float* p = xyz1 + ((size_t)b * N_ + n0 + mrow) * 3;
    v2f a;
    if (lane < 16) { a.x = -2.0f * p[0]; a.y = -2.0f * p[1]; }
    else           { a.x = -2.0f * p[2]; a.y = 1.0f; }

    // per-lane running top-3 for the 8 C-matrix rows this lane holds
    float d0[8], d1[8], d2[8];
    int   i0[8], i1[8], i2[8];
#pragma unroll
    for (int r = 0; r < 8; ++r) {
        d0[r] = d1[r] = d2[r] = 3.0e38f;
        i0[r] = i1[r] = i2[r] = 0;
    }

    const v2f* bbase = wsB + (((size_t)b * MTILES) << 5) + lane;
    const int  jcol  = lane & 15;

#pragma unroll 2
    for (int t = 0; t < MTILES; ++t) {
        v2f bt = bbase[(size_t)t << 5];          // one b64 load / lane / tile
        v8f cz = {};
        // D = A(16x4) * B(4x16) + 0  -> score = |q|^2 - 2 p.q
        v8f c = __builtin_amdgcn_wmma_f32_16x16x4_f32(
            false, a, false, bt, (short)0, cz, false, false);
        int jidx = (t << 4) + jcol;
#pragma unroll
        for (int r = 0; r < 8; ++r)
            top3_insert(c[r], jidx, d0[r], d1[r], d2[r], i0[r], i1[r], i2[r]);
    }

    // ---- merge the 16 per-lane top-3 lists within each 16-lane half ----
#pragma unroll
    for (int mask = 1; mask <= 8; mask <<= 1) {
#pragma unroll
        for (int r = 0; r < 8; ++r) {
            float od0 = __shfl_xor(d0[r], mask, 32);
            int   oi0 = __shfl_xor(i0[r], mask, 32);
            float od1 = __shfl_xor(d1[r], mask, 32);
            int   oi1 = __shfl_xor(i1[r], mask, 32);
            float od2 = __shfl_xor(d2[r], mask, 32);
            int   oi2 = __shfl_xor(i2[r], mask, 32);
            top3_insert(od0, oi0, d0[r], d1[r], d2[r], i0[r], i1[r], i2[r]);
            top3_insert(od1, oi1, d0[r], d1[r], d2[r], i0[r], i1[r], i2[r]);
            top3_insert(od2, oi2, d0[r], d1[r], d2[r], i0[r], i1[r], i2[r]);
        }
    }

    // ---- weights + feature gather: each lane handles 2 of 64 channels ----
    const float* fbase = feat2 + (size_t)b * M_ * C_;
    float*       obase = out   + ((size_t)b * N_ + n0) * C_;
    const int    ch    = lane * 2;

#pragma unroll
    for (int m = 0; m < 16; ++m) {
        const int src = (m & 8) ? 16 : 0;      // half-wave that owns row m
        const int r   = m & 7;
        float s0 = __shfl(d0[r], src, 32);  int j0 = __shfl(i0[r], src, 32);
        float s1 = __shfl(d1[r], src, 32);  int j1 = __shfl(i1[r], src, 32);
        float s2 = __shfl(d2[r], src, 32);  int j2 = __shfl(i2[r], src, 32);

        // add back |p|^2 (row-constant, was dropped from the GEMM)
        const float* pp = xyz1 + ((size_t)b * N_ + n0 + m) * 3;
        float px = pp[0], py = pp[1], pz = pp[2];
        float pn = px * px + py * py + pz * pz;

        float w0 = 1.0f / (fabsf(s0 + pn) + EPS_);
        float w1 = 1.0f / (fabsf(s1 + pn) + EPS_);
        float w2 = 1.0f / (fabsf(s2 + pn) + EPS_);
        float inv = 1.0f / (w0 + w1 + w2);
        w0 *= inv; w1 *= inv; w2 *= inv;

        v2f f0 = *(const v2f*)(fbase + (size_t)j0 * C_ + ch);
        v2f f1 = *(const v2f*)(fbase + (size_t)j1 * C_ + ch);
        v2f f2 = *(const v2f*)(fbase + (size_t)j2 * C_ + ch);
        v2f o;
        o.x = w0 * f0.x + w1 * f1.x + w2 * f2.x;
        o.y = w0 * f0.y + w1 * f1.y + w2 * f2.y;
        *(v2f*)(obase + (size_t)m * C_ + ch) = o;
    }
}

// ---------------------------------------------------------------------------
extern "C" void kernel_launch(void* const* d_in, const int* in_sizes, int n_in,
                              void* d_out, int out_size, void* d_ws, size_t ws_size,
                              hipStream_t stream) {
    const float* xyz1  = (const float*)d_in[0];   // [B,N,3]
    const float* xyz2  = (const float*)d_in[1];   // [B,M,3]
    const float* feat2 = (const float*)d_in[2];   // [B,M,C]
    float*       out   = (float*)d_out;           // [B,N,C]
    v2f*         wsB   = (v2f*)d_ws;              // B*M*2 v2f = 256 KB

    (void)in_sizes; (void)n_in; (void)out_size; (void)ws_size;

    IntpFeat_prep_kernel<<<(B_ * M_ + 255) / 256, 256, 0, stream>>>(xyz2, wsB);

    const int waves = (B_ * N_) / 16;             // 4096
    IntpFeat_knn_kernel<<<waves / WPB, WPB * 32, 0, stream>>>(xyz1, feat2, wsB, out);
}